// ContinuousSoftmax_2757369004821
// MI455X (gfx1250) — compile-verified
//
#include <hip/hip_runtime.h>
#include <hip/hip_bf16.h>

// ContinuousSoftmax: r[n,m] = N(Mu_n - mu_m ; 0, Sigma_n + S_m)
// Store-bandwidth bound: ~56 MB traffic -> ~2.4 us floor at 23.3 TB/s.
// CDNA5/gfx1250 paths: global_load_async_to_lds_b128/b32 + s_wait_asynccnt
// (ASYNCcnt discipline), global_prefetch_b8, wave32 layout (wave-per-row,
// 4 columns per lane, coalesced non-temporal global_store_b128).
// WMMA intentionally unused: per-(n,m) work is a 2x2 SPD inverse inside exp();
// no 16x16xK matmul structure exists and the kernel is at the HBM store roof.

#define BLOCK 256
#define WPB   (BLOCK / 32)   // waves per block (wave32)
#define MAX_M 1024           // LDS: 6*MAX_M floats = 24 KB / block
#define MAX_BLOCKS 2048      // amortize LDS staging over >= 8 rows per wave

typedef __attribute__((ext_vector_type(4))) float v4f;  // native vector for NT store

__global__ __launch_bounds__(BLOCK)
void continuous_softmax_kernel(const float* __restrict__ theta,
                               const float* __restrict__ basis_mu,
                               const float* __restrict__ basis_sigma,
                               float* __restrict__ out,
                               int N, int M) {
    alignas(16) __shared__ float s_mu[2 * MAX_M];   // [M][2]
    alignas(16) __shared__ float s_sg[4 * MAX_M];   // [M][2][2]

    const int tid = threadIdx.x;

    // ---- Stage basis tables into LDS with CDNA5 async global->LDS DMA ----
    {
        // addrspace(3) offset == low 32 bits of the generic pointer on amdgcn
        const unsigned mu_lds = (unsigned)(uintptr_t)(void*)s_mu;
        const unsigned sg_lds = (unsigned)(uintptr_t)(void*)s_sg;

        const int mu_floats = 2 * M;
        const int sg_floats = 4 * M;

        for (int i = tid; i < (mu_floats >> 2); i += BLOCK) {
            unsigned lds = mu_lds + (unsigned)(i * 16);
            unsigned off = (unsigned)(i * 16);
            asm volatile("global_load_async_to_lds_b128 %0, %1, %2"
                         :: "v"(lds), "v"(off), "s"(basis_mu) : "memory");
        }
        for (int i = ((mu_floats >> 2) << 2) + tid; i < mu_floats; i += BLOCK) {
            unsigned lds = mu_lds + (unsigned)(i * 4);
            unsigned off = (unsigned)(i * 4);
            asm volatile("global_load_async_to_lds_b32 %0, %1, %2"
                         :: "v"(lds), "v"(off), "s"(basis_mu) : "memory");
        }
        for (int i = tid; i < (sg_floats >> 2); i += BLOCK) {
            unsigned lds = sg_lds + (unsigned)(i * 16);
            unsigned off = (unsigned)(i * 16);
            asm volatile("global_load_async_to_lds_b128 %0, %1, %2"
                         :: "v"(lds), "v"(off), "s"(basis_sigma) : "memory");
        }
        asm volatile("s_wait_asynccnt 0" ::: "memory");
    }
    __syncthreads();

    const int lane = tid & 31;
    const int wave = tid >> 5;
    const int wid  = blockIdx.x * WPB + wave;          // global wave id
    const int wstr = (int)gridDim.x * WPB;             // wave grid stride

    const int  M4   = (M + 3) >> 2;                    // groups of 4 columns
    const bool vec4 = ((M & 3) == 0);

    const float kNegHalfLog2e = -0.72134752044448170368f; // -0.5 * log2(e)
    const float kLog2Inv2Pi   = -2.65149612947231879804f; // log2(1/(2*pi))

    for (int q = lane; q < M4; q += 32) {
        const int mb = q << 2;
        // Hoist this lane's 4 basis entries into registers (reused for all rows)
        float bm0[4], bm1[4], g00[4], g01[4], g10[4], g11[4];
#pragma unroll
        for (int k = 0; k < 4; ++k) {
            int m = mb + k; if (m >= M) m = M - 1;     // clamp; tail guarded at store
            bm0[k] = s_mu[2 * m + 0];
            bm1[k] = s_mu[2 * m + 1];
            g00[k] = s_sg[4 * m + 0];
            g01[k] = s_sg[4 * m + 1];
            g10[k] = s_sg[4 * m + 2];
            g11[k] = s_sg[4 * m + 3];
        }

        for (int n = wid; n < N; n += wstr) {
            const int nu = __builtin_amdgcn_readfirstlane(n);  // wave-uniform -> SMEM
            const float* tp = theta + (size_t)nu * 6u;
            if (nu + wstr < N)                                  // global_prefetch_b8
                __builtin_prefetch(theta + (size_t)(nu + wstr) * 6u, 0, 0);

            const float e0 = tp[0], e1 = tp[1];
            const float p00 = -2.0f * tp[2], p01 = -2.0f * tp[3];
            const float p10 = -2.0f * tp[4], p11 = -2.0f * tp[5];
            const float invd = __builtin_amdgcn_rcpf(p00 * p11 - p01 * p10);
            const float s00 = p11 * invd;
            const float s11 = p00 * invd;
            const float s01 = -0.5f * (p01 + p10) * invd;      // symmetrized Pinv
            const float mu0 = s00 * e0 + s01 * e1;
            const float mu1 = s01 * e0 + s11 * e1;

            float rv[4];
#pragma unroll
            for (int k = 0; k < 4; ++k) {
                const float c00 = s00 + g00[k];
                const float c01 = s01 + g01[k];
                const float c10 = s01 + g10[k];
                const float c11 = s11 + g11[k];
                const float det = c00 * c11 - c01 * c10;
                const float dx  = mu0 - bm0[k];
                const float dy  = mu1 - bm1[k];
                // rsq once; invdet = rsq^2 (saves a v_rcp per element)
                const float rsq  = __builtin_amdgcn_rsqf(det);
                const float qf   = (c11 * dx * dx + c00 * dy * dy
                                    - (c01 + c10) * dx * dy) * (rsq * rsq);
                // exp(-q/2)/(2*pi) = exp2(q * -0.5*log2(e) + log2(1/(2*pi)))
                rv[k] = __builtin_amdgcn_exp2f(
                            __builtin_fmaf(qf, kNegHalfLog2e, kLog2Inv2Pi)) * rsq;
            }

            float* orow = out + (size_t)nu * (size_t)M;
            if (vec4 && (mb + 3 < M)) {
                v4f v;
                v.x = rv[0]; v.y = rv[1]; v.z = rv[2]; v.w = rv[3];
                // streaming write-once output: non-temporal b128 store
                __builtin_nontemporal_store(v,
                    reinterpret_cast<v4f*>(orow + mb));
            } else {
#pragma unroll
                for (int k = 0; k < 4; ++k)
                    if (mb + k < M) orow[mb + k] = rv[k];
            }
        }
    }
}

extern "C" void kernel_launch(void* const* d_in, const int* in_sizes, int n_in,
                              void* d_out, int out_size, void* d_ws, size_t ws_size,
                              hipStream_t stream) {
    const float* theta       = (const float*)d_in[0];  // [N,6]
    const float* basis_mu    = (const float*)d_in[1];  // [M,2]
    const float* basis_sigma = (const float*)d_in[2];  // [M,2,2]
    float* out = (float*)d_out;                        // [N,M]

    const int N = in_sizes[0] / 6;
    const int M = in_sizes[1] / 2;                     // 100 (<= MAX_M)

    int blocks = (N + WPB - 1) / WPB;
    if (blocks > MAX_BLOCKS) blocks = MAX_BLOCKS;      // amortize LDS staging
    if (blocks < 1) blocks = 1;

    continuous_softmax_kernel<<<blocks, BLOCK, 0, stream>>>(
        theta, basis_mu, basis_sigma, out, N, M);
}